// QueryBallPoint_9732395892839
// MI455X (gfx1250) — compile-verified
//
#include <hip/hip_runtime.h>

// PointNet++ ball query on gfx1250 (MI455X).
// d2(p,c) = |p|^2 - 2 p.c + |c|^2 computed as a 16x16 tile per
// V_WMMA_F32_16X16X4_F32 (A = 16 points x K4, B = K4 x 16 centroids,
// C preloaded with |c|^2). Ordered first-64 selection via per-lane masks +
// ds_bpermute prefix exchange between partner lanes (wave32).

typedef float v2f __attribute__((ext_vector_type(2)));
typedef float v8f __attribute__((ext_vector_type(8)));

#define RADIUS2  0.01f
#define NSAMPLE  64
#define NPTS     16384
#define NCENT    4096
#define NBATCH   4
#define CHUNKS   (NPTS / 16)

__device__ __forceinline__ int bperm(int srcLane, int v) {
  // wave32: ds_bpermute uses byte addr bits [6:2] as lane index
  return __builtin_amdgcn_ds_bpermute(srcLane << 2, v);
}
__device__ __forceinline__ unsigned ballot32(bool p) {
  return __builtin_amdgcn_ballot_w32(p);
}

__global__ __launch_bounds__(256)
void BallQueryWMMA_kernel(const float* __restrict__ xyz1,
                          const float* __restrict__ xyz2,
                          int* __restrict__ out) {
  const int lane  = threadIdx.x & 31;
  const int wave  = blockIdx.x * (blockDim.x >> 5) + (threadIdx.x >> 5);
  const int b     = wave >> 8;        // 256 centroid tiles per batch
  const int mtile = wave & 255;
  const int half  = (lane < 16) ? 0 : 1;      // partner-half id
  const int cidx  = mtile * 16 + (lane & 15); // centroid within batch
  const int g     = b * NCENT + cidx;         // global centroid id

  const float* pts = xyz1 + (size_t)b * NPTS * 3;
  const float* ctr = xyz2 + (size_t)g * 3;

  // --- B operand: column n = centroid (-2cx,-2cy,-2cz,1); rows split 0,2|1,3
  const float cx = ctr[0], cy = ctr[1], cz = ctr[2];
  const float cnorm = fmaf(cz, cz, fmaf(cy, cy, cx * cx));
  v2f bmat;
  bmat[0] = half ? (-2.0f * cz) : (-2.0f * cx);  // K=2 | K=0
  bmat[1] = half ?        1.0f  : (-2.0f * cy);  // K=3 | K=1
  v8f cacc;
#pragma unroll
  for (int k = 0; k < 8; ++k) cacc[k] = cnorm;   // C[m][n] = |c_n|^2

  int* __restrict__ idxOut = out + (size_t)g * NSAMPLE;
  int* __restrict__ cntOut = out + (size_t)NBATCH * NCENT * NSAMPLE;

  const int rowbase = half ? 8 : 0;  // which 8 rows of the 16-pt chunk
  const int pl = lane & 15;          // which point of the chunk this lane loads
  int cnt = 0;                       // in-radius count for my centroid
  int firstIdx = -1;                 // index written to slot 0 (if by me)

  // software-pipelined point coords for chunk 0
  float px = pts[pl * 3 + 0], py = pts[pl * 3 + 1], pz = pts[pl * 3 + 2];

  for (int chunk = 0; chunk < CHUNKS; ++chunk) {
    const float x = px, y = py, z = pz;
    // branchless prefetch of next chunk's point (clamped: never OOB, last
    // iteration redundantly reloads in-bounds data). Unconditional load with
    // results consumed only next iteration lets the scheduler sink the wait
    // past the WMMA + min-reduce below.
    {
      const int nc = (chunk + 1 < CHUNKS) ? (chunk + 1) : (CHUNKS - 1);
      const float* np = pts + ((size_t)nc * 16 + pl) * 3;
      px = np[0]; py = np[1]; pz = np[2];
    }
    const float pn = fmaf(z, z, fmaf(y, y, x * x));

    // --- A operand: row m = point (x,y,z,|p|^2); rows split K 0,2 | 1,3
    v2f amat;
    amat[0] = half ? z  : x;   // K=2 | K=0
    amat[1] = half ? pn : y;   // K=3 | K=1

    // D[m][n] = |p_m|^2 - 2 p_m . c_n + |c_n|^2  (16x16 tile, 256 pairs)
    v8f d = __builtin_amdgcn_wmma_f32_16x16x4_f32(
        false, amat, false, bmat, (short)0, cacc, false, false);

    // cheap uniform "any hit in wave" test: min-reduce my 8 distances
    const float dmin =
        fminf(fminf(fminf(d[0], d[1]), fminf(d[2], d[3])),
              fminf(fminf(d[4], d[5]), fminf(d[6], d[7])));
    if (ballot32(dmin < RADIUS2) != 0u) {  // uniform branch; EXEC stays full
      unsigned m = 0;
#pragma unroll
      for (int i = 0; i < 8; ++i) m |= (d[i] < RADIUS2) ? (1u << i) : 0u;
      const int pc = __popc((int)m);
      // partner-lane exchange: low half's count is the high half's prefix
      const int lowCnt  = bperm(lane & 15, pc);
      const int highCnt = bperm((lane & 15) + 16, pc);
      int slot = cnt + (half ? lowCnt : 0);
      if (m != 0) {
        const int pbase = chunk * 16 + rowbase;
#pragma unroll
        for (int i = 0; i < 8; ++i) {
          if (m & (1u << i)) {
            if (slot < NSAMPLE) {
              idxOut[slot] = pbase + i;
              if (slot == 0) firstIdx = pbase + i;
            }
            ++slot;
          }
        }
      }
      cnt += lowCnt + highCnt;  // identical in both partner lanes
      // uniform early exit once every centroid in the tile has 64 hits
      if (ballot32(cnt >= NSAMPLE) == 0xFFFFFFFFu) break;
    }
  }

  // --- epilogue: padding + counts -------------------------------------
  // slot 0's value lives in exactly one of the two partner lanes (or none)
  const int otherFirst = bperm(lane ^ 16, firstIdx);
  int first = (firstIdx > otherFirst) ? firstIdx : otherFirst;
  if (first < 0) first = 0;  // no in-radius point -> pad with 0
  const int cntC = (cnt < NSAMPLE) ? cnt : NSAMPLE;
  // partner lanes interleave the padding writes
  for (int s = cntC + half; s < NSAMPLE; s += 2) idxOut[s] = first;
  if (!half) cntOut[g] = cntC;
}

extern "C" void kernel_launch(void* const* d_in, const int* in_sizes, int n_in,
                              void* d_out, int out_size, void* d_ws, size_t ws_size,
                              hipStream_t stream) {
  (void)in_sizes; (void)n_in; (void)out_size; (void)d_ws; (void)ws_size;
  const float* xyz1 = (const float*)d_in[0];  // (4,16384,3) f32
  const float* xyz2 = (const float*)d_in[1];  // (4,4096,3)  f32
  int* out = (int*)d_out;                     // idx (4,4096,64) then cnt (4,4096)
  // 1024 waves total = NBATCH * (NCENT/16); 8 waves (256 thr) per block
  dim3 grid((NBATCH * (NCENT / 16)) / 8), block(256);
  hipLaunchKernelGGL(BallQueryWMMA_kernel, grid, block, 0, stream,
                     xyz1, xyz2, out);
}